// PointnetPlaneVotes_4209067950526
// MI455X (gfx1250) — compile-verified
//
#include <hip/hip_runtime.h>
#include <math.h>

// ---------------- problem constants (from reference) ----------------
#define BQ    8
#define NPTS  4096
#define CF    256
#define SQ    512
#define NSAMP 64
#define KIN   259          // 3 + C
#define H     128          // hidden width of all MLP layers
#define RAD   0.3f
#define EPS_  1e-5f

#define NTILE_LOCAL (BQ*SQ)        // 4096 tiles of 64 grouped points
#define NTILE_GLOB  (BQ*(NPTS/64)) // 512 tiles of 64 raw points

typedef float v2f __attribute__((ext_vector_type(2)));
typedef float v8f __attribute__((ext_vector_type(8)));

#define XS_STRIDE 132   // LDS K-stride (even, 132 % 64 == 4 -> conflict-free B loads)

// =====================================================================
// 0) zero the stats accumulators (6 layers x [sum|sumsq|a|d] x 128)
// =====================================================================
__global__ void zero_stats_kernel(float* stats) {
    int i = blockIdx.x * blockDim.x + threadIdx.x;
    if (i < 6 * 4 * H) stats[i] = 0.0f;
}

// =====================================================================
// 1) ball query: one wave per centroid; first NSAMP in-radius indices
//    in ascending index order (== reference's sort-of-candidates),
//    remainder filled with the first hit. Also writes new_xyz to d_out.
// =====================================================================
__global__ __launch_bounds__(256) void bq_kernel(const float* __restrict__ xyz,
                                                 const int*   __restrict__ inds,
                                                 int*   __restrict__ idxbuf,
                                                 float* __restrict__ out_newxyz) {
    const int wid  = blockIdx.x * 8 + (threadIdx.x >> 5);
    const int lane = threadIdx.x & 31;
    const int b = wid >> 9;          // / 512
    const int s = wid & 511;
    const int ci = inds[b * SQ + s];
    const float* cp = xyz + ((size_t)b * NPTS + ci) * 3;
    const float cx = cp[0], cy = cp[1], cz = cp[2];
    if (lane < 3) out_newxyz[((size_t)b * SQ + s) * 3 + lane] = cp[lane];

    const float r2 = RAD * RAD;
    int* my = idxbuf + (size_t)wid * NSAMP;
    int total = 0, first = -1;
    for (int base = 0; base < NPTS && total < NSAMP; base += 32) {
        const int n = base + lane;
        const float* pp = xyz + ((size_t)b * NPTS + n) * 3;
        const float dx = pp[0] - cx, dy = pp[1] - cy, dz = pp[2] - cz;
        const bool pred = (dx * dx + dy * dy + dz * dz) < r2;
        const unsigned mask = __builtin_amdgcn_ballot_w32(pred);
        if (pred) {
            const int pos = total + __builtin_popcount(mask & ((1u << lane) - 1u));
            if (pos < NSAMP) my[pos] = n;
        }
        if (first < 0 && mask != 0u) first = base + __builtin_ctz(mask);
        total += __builtin_popcount(mask);
    }
    if (first < 0) first = 0;  // cannot happen (centroid is its own neighbor)
    for (int p = total + lane; p < NSAMP; p += 32) my[p] = first;
}

// =====================================================================
// WMMA GEMM core: each block computes a 128(M) x 64(N) fp32 tile.
// Wave w: M rows [16w,16w+16), 4 N-subtiles, K stepped by 4 via
// V_WMMA_F32_16X16X4_F32.  A (16x4): lane%16 = M row, VGPR pair holds
// K = k0+2*(lane/16) .. +1.  B (4x16): lane%16 = N col, same K halves.
// =====================================================================

// ---- layer 0 (K=259), input assembled from gather (local) or raw (glob)
__global__ __launch_bounds__(256) void conv_first_kernel(
    const float* __restrict__ W,        // [128 x 259]
    const float* __restrict__ xyz,
    const float* __restrict__ feat,     // [B x C x N]
    const int*   __restrict__ inds,
    const int*   __restrict__ idxbuf,   // [tiles x 64]
    float* __restrict__ yout,           // [tiles x 128 x 64] raw (pre-BN)
    float* __restrict__ gsum, float* __restrict__ gsq,
    int is_local)
{
    __shared__ float Xs[64 * XS_STRIDE];
    __shared__ int   pj[64];
    __shared__ float ctr[3];
    __shared__ float s_sum[H], s_sq[H];

    const int tid  = threadIdx.x;
    const int tile = blockIdx.x;
    const int b = is_local ? (tile >> 9) : (tile >> 6);

    if (tid < 64) {
        pj[tid] = is_local ? idxbuf[(size_t)tile * 64 + tid]
                           : ((tile & 63) << 6) + tid;
    }
    if (is_local && tid < 3) {
        const int s  = tile & 511;
        const int ci = inds[b * SQ + s];
        ctr[tid] = xyz[((size_t)b * NPTS + ci) * 3 + tid];
    }
    if (tid < H) { s_sum[tid] = 0.0f; s_sq[tid] = 0.0f; }
    __syncthreads();

    const int w     = tid >> 5;
    const int lane  = tid & 31;
    const int m     = (w << 4) + (lane & 15);
    const int khalf = (lane >> 4) << 1;  // 0 or 2
    const float invR = 1.0f / RAD;

    v8f acc[4];
    #pragma unroll
    for (int t = 0; t < 4; ++t)
        acc[t] = (v8f){0.f, 0.f, 0.f, 0.f, 0.f, 0.f, 0.f, 0.f};

    // two K-phases of 130 (fits LDS; covers K=259 with zero padding)
    for (int phase = 0; phase < 2; ++phase) {
        const int kbase = phase * 130;
        for (int e = tid; e < 64 * XS_STRIDE; e += 256) {
            const int j = e / XS_STRIDE, k = e - j * XS_STRIDE;
            const int kg = kbase + k;
            float v = 0.0f;
            if (k < 130 && kg < KIN) {
                const int p = pj[j];
                if (kg < 3) {
                    const float x = xyz[((size_t)b * NPTS + p) * 3 + kg];
                    v = is_local ? (x - ctr[kg]) * invR : x;
                } else {
                    v = feat[((size_t)b * CF + (kg - 3)) * NPTS + p];
                }
            }
            Xs[e] = v;
        }
        __syncthreads();

        for (int k0 = 0; k0 < 132; k0 += 4) {
            const int kb  = k0 + khalf;
            const int kgA = kbase + kb;
            const float a0 = (kgA     < KIN) ? W[m * KIN + kgA]     : 0.0f;
            const float a1 = (kgA + 1 < KIN) ? W[m * KIN + kgA + 1] : 0.0f;
            const v2f A = {a0, a1};
            #pragma unroll
            for (int t = 0; t < 4; ++t) {
                const v2f Bv = *(const v2f*)&Xs[(((t << 4) + (lane & 15)) * XS_STRIDE) + kb];
                acc[t] = __builtin_amdgcn_wmma_f32_16x16x4_f32(
                    false, A, false, Bv, (short)0, acc[t], false, false);
            }
        }
        __syncthreads();
    }

    // store raw y + accumulate per-channel sum / sum^2
    const int jbase = lane & 15;
    const int ohalf = (lane >> 4) << 3;  // 0 or 8
    #pragma unroll
    for (int v = 0; v < 8; ++v) {
        const int o = (w << 4) + ohalf + v;
        float part = 0.0f, psq = 0.0f;
        #pragma unroll
        for (int t = 0; t < 4; ++t) {
            const float val = acc[t][v];
            yout[((size_t)tile * H + o) * 64 + (t << 4) + jbase] = val;
            part += val; psq += val * val;
        }
        atomicAdd(&s_sum[o], part);
        atomicAdd(&s_sq[o], psq);
    }
    __syncthreads();
    if (tid < H) { atomicAdd(&gsum[tid], s_sum[tid]); atomicAdd(&gsq[tid], s_sq[tid]); }
}

// ---- layers 1/2 (K=128): BN(a,d)+relu folded into LDS staging
__global__ __launch_bounds__(256) void conv_mid_kernel(
    const float* __restrict__ W,        // [128 x 128]
    const float* __restrict__ yin,      // [tiles x 128 x 64] raw (pre-BN)
    const float* __restrict__ bnA, const float* __restrict__ bnD,
    float* __restrict__ yout,
    float* __restrict__ gsum, float* __restrict__ gsq)
{
    __shared__ float Xs[64 * XS_STRIDE];
    __shared__ float s_sum[H], s_sq[H];

    const int tid  = threadIdx.x;
    const int tile = blockIdx.x;
    if (tid < H) { s_sum[tid] = 0.0f; s_sq[tid] = 0.0f; }

    const float* src = yin + (size_t)tile * (H * 64);
    for (int e = tid; e < H * 64; e += 256) {
        const int k = e >> 6, j = e & 63;
        float v = bnA[k] * src[e] + bnD[k];
        v = v > 0.0f ? v : 0.0f;
        Xs[j * XS_STRIDE + k] = v;
    }
    __syncthreads();

    const int w     = tid >> 5;
    const int lane  = tid & 31;
    const int m     = (w << 4) + (lane & 15);
    const int khalf = (lane >> 4) << 1;

    v8f acc[4];
    #pragma unroll
    for (int t = 0; t < 4; ++t)
        acc[t] = (v8f){0.f, 0.f, 0.f, 0.f, 0.f, 0.f, 0.f, 0.f};

    for (int k0 = 0; k0 < H; k0 += 4) {
        const int kb = k0 + khalf;
        const v2f A = *(const v2f*)&W[m * H + kb];
        #pragma unroll
        for (int t = 0; t < 4; ++t) {
            const v2f Bv = *(const v2f*)&Xs[(((t << 4) + (lane & 15)) * XS_STRIDE) + kb];
            acc[t] = __builtin_amdgcn_wmma_f32_16x16x4_f32(
                false, A, false, Bv, (short)0, acc[t], false, false);
        }
    }

    const int jbase = lane & 15;
    const int ohalf = (lane >> 4) << 3;
    #pragma unroll
    for (int v = 0; v < 8; ++v) {
        const int o = (w << 4) + ohalf + v;
        float part = 0.0f, psq = 0.0f;
        #pragma unroll
        for (int t = 0; t < 4; ++t) {
            const float val = acc[t][v];
            yout[((size_t)tile * H + o) * 64 + (t << 4) + jbase] = val;
            part += val; psq += val * val;
        }
        atomicAdd(&s_sum[o], part);
        atomicAdd(&s_sq[o], psq);
    }
    __syncthreads();
    if (tid < H) { atomicAdd(&gsum[tid], s_sum[tid]); atomicAdd(&gsq[tid], s_sq[tid]); }
}

// =====================================================================
// BN stats -> per-channel affine (a = g*rsqrt(var+eps), d = be - mu*a)
// =====================================================================
__global__ void bn_finalize_kernel(const float* __restrict__ gsum,
                                   const float* __restrict__ gsq,
                                   const float* __restrict__ g,
                                   const float* __restrict__ be,
                                   float cnt_inv,
                                   float* __restrict__ bnA, float* __restrict__ bnD) {
    const int o = threadIdx.x;
    if (o < H) {
        const float mu  = gsum[o] * cnt_inv;
        const float var = gsq[o] * cnt_inv - mu * mu;
        const float inv = rsqrtf(var + EPS_);
        const float a   = g[o] * inv;
        bnA[o] = a;
        bnD[o] = be[o] - mu * a;
    }
}

// =====================================================================
// pooling: local -> out channels [0,128), glob -> [128,256) broadcast
// =====================================================================
__global__ void pool_local_kernel(const float* __restrict__ y,
                                  const float* __restrict__ bnA,
                                  const float* __restrict__ bnD,
                                  float* __restrict__ out) {
    const int t = blockIdx.x, o = threadIdx.x;
    const float* p = y + ((size_t)t * H + o) * 64;
    const float a = bnA[o], d = bnD[o];
    float mx = -3.0e38f;
    for (int j = 0; j < 64; ++j) mx = fmaxf(mx, a * p[j] + d);
    mx = fmaxf(mx, 0.0f);  // relu commutes with max
    const int b = t >> 9, s = t & 511;
    out[(size_t)BQ * SQ * 3 + ((size_t)b * 256 + o) * SQ + s] = mx;
}

__global__ void pool_glob_kernel(const float* __restrict__ y,
                                 const float* __restrict__ bnA,
                                 const float* __restrict__ bnD,
                                 float* __restrict__ out) {
    const int b = blockIdx.x, o = threadIdx.x;
    const float a = bnA[o], d = bnD[o];
    float mx = -3.0e38f;
    for (int tt = 0; tt < 64; ++tt) {
        const float* p = y + (((size_t)(b * 64 + tt)) * H + o) * 64;
        for (int j = 0; j < 64; ++j) mx = fmaxf(mx, a * p[j] + d);
    }
    mx = fmaxf(mx, 0.0f);
    float* dst = out + (size_t)BQ * SQ * 3 + ((size_t)b * 256 + 128 + o) * SQ;
    for (int s = 0; s < SQ; ++s) dst[s] = mx;
}

// =====================================================================
extern "C" void kernel_launch(void* const* d_in, const int* in_sizes, int n_in,
                              void* d_out, int out_size, void* d_ws, size_t ws_size,
                              hipStream_t stream) {
    const float* xyz  = (const float*)d_in[0];
    const float* feat = (const float*)d_in[1];
    const int*   inds = (const int*)d_in[2];
    const float* w10 = (const float*)d_in[3];
    const float* g10 = (const float*)d_in[4];
    const float* be10= (const float*)d_in[5];
    const float* w11 = (const float*)d_in[6];
    const float* g11 = (const float*)d_in[7];
    const float* be11= (const float*)d_in[8];
    const float* w12 = (const float*)d_in[9];
    const float* g12 = (const float*)d_in[10];
    const float* be12= (const float*)d_in[11];
    const float* w20 = (const float*)d_in[12];
    const float* g20 = (const float*)d_in[13];
    const float* be20= (const float*)d_in[14];
    const float* w21 = (const float*)d_in[15];
    const float* g21 = (const float*)d_in[16];
    const float* be21= (const float*)d_in[17];
    const float* w22 = (const float*)d_in[18];
    const float* g22 = (const float*)d_in[19];
    const float* be22= (const float*)d_in[20];

    float* out = (float*)d_out;
    char*  ws  = (char*)d_ws;

    // workspace layout
    int*   idxbuf = (int*)ws;                                   // 1 MiB
    float* stats  = (float*)(ws + (1u << 20));                  // 12 KiB
    float* yA     = (float*)(ws + (2u << 20));                  // 128 MiB
    float* yB     = yA  + (size_t)NTILE_LOCAL * H * 64;         // 128 MiB
    float* ygA    = yB  + (size_t)NTILE_LOCAL * H * 64;         // 16 MiB
    float* ygB    = ygA + (size_t)NTILE_GLOB  * H * 64;         // 16 MiB

    // per-layer stats blocks: [sum(128) | sumsq(128) | a(128) | d(128)]
    #define SUMP(L)  (stats + (L) * 4 * H)
    #define SQP(L)   (stats + (L) * 4 * H + H)
    #define AP(L)    (stats + (L) * 4 * H + 2 * H)
    #define DP(L)    (stats + (L) * 4 * H + 3 * H)

    const float invCntL = 1.0f / (float)((size_t)BQ * SQ * NSAMP);  // 262144
    const float invCntG = 1.0f / (float)((size_t)BQ * NPTS);        // 32768

    zero_stats_kernel<<<12, 256, 0, stream>>>(stats);
    bq_kernel<<<NTILE_LOCAL / 8, 256, 0, stream>>>(xyz, inds, idxbuf, out);

    // ---- local branch: 259 -> 128 -> 128 -> 128, tiles of (b,s) ----
    conv_first_kernel<<<NTILE_LOCAL, 256, 0, stream>>>(w10, xyz, feat, inds, idxbuf,
                                                       yA, SUMP(0), SQP(0), 1);
    bn_finalize_kernel<<<1, H, 0, stream>>>(SUMP(0), SQP(0), g10, be10, invCntL, AP(0), DP(0));
    conv_mid_kernel<<<NTILE_LOCAL, 256, 0, stream>>>(w11, yA, AP(0), DP(0),
                                                     yB, SUMP(1), SQP(1));
    bn_finalize_kernel<<<1, H, 0, stream>>>(SUMP(1), SQP(1), g11, be11, invCntL, AP(1), DP(1));
    conv_mid_kernel<<<NTILE_LOCAL, 256, 0, stream>>>(w12, yB, AP(1), DP(1),
                                                     yA, SUMP(2), SQP(2));
    bn_finalize_kernel<<<1, H, 0, stream>>>(SUMP(2), SQP(2), g12, be12, invCntL, AP(2), DP(2));
    pool_local_kernel<<<NTILE_LOCAL, H, 0, stream>>>(yA, AP(2), DP(2), out);

    // ---- global branch: same MLP over all N points, tiles of 64 ----
    conv_first_kernel<<<NTILE_GLOB, 256, 0, stream>>>(w20, xyz, feat, inds, idxbuf,
                                                      ygA, SUMP(3), SQP(3), 0);
    bn_finalize_kernel<<<1, H, 0, stream>>>(SUMP(3), SQP(3), g20, be20, invCntG, AP(3), DP(3));
    conv_mid_kernel<<<NTILE_GLOB, 256, 0, stream>>>(w21, ygA, AP(3), DP(3),
                                                    ygB, SUMP(4), SQP(4));
    bn_finalize_kernel<<<1, H, 0, stream>>>(SUMP(4), SQP(4), g21, be21, invCntG, AP(4), DP(4));
    conv_mid_kernel<<<NTILE_GLOB, 256, 0, stream>>>(w22, ygB, AP(4), DP(4),
                                                    ygA, SUMP(5), SQP(5));
    bn_finalize_kernel<<<1, H, 0, stream>>>(SUMP(5), SQP(5), g22, be22, invCntG, AP(5), DP(5));
    pool_glob_kernel<<<BQ, H, 0, stream>>>(ygA, AP(5), DP(5), out);

    #undef SUMP
    #undef SQP
    #undef AP
    #undef DP
}